// RiemannianFeatureEncoder_57475252355079
// MI455X (gfx1250) — compile-verified
//
#include <hip/hip_runtime.h>
#include <hip/hip_bf16.h>
#include <math.h>

typedef __attribute__((ext_vector_type(2))) float v2f;
typedef __attribute__((ext_vector_type(8))) float v8f;

#define DD 8

// ---------------- geometry helpers (C = 1) ----------------

__device__ __forceinline__ float dot8(const float* a, const float* b) {
  float s = 0.f;
#pragma unroll
  for (int j = 0; j < DD; ++j) s += a[j] * b[j];
  return s;
}

__device__ __forceinline__ void mobius_add8(const float* x, const float* y, float* o) {
  float xy = 0.f, nx = 0.f, ny = 0.f;
#pragma unroll
  for (int j = 0; j < DD; ++j) { xy += x[j]*y[j]; nx += x[j]*x[j]; ny += y[j]*y[j]; }
  float a = 1.f + 2.f*xy + ny;
  float b = 1.f - nx;
  float rden = 1.f / (1.f + 2.f*xy + nx*ny + 1e-10f);
#pragma unroll
  for (int j = 0; j < DD; ++j) o[j] = (a*x[j] + b*y[j]) * rden;
}

__device__ __forceinline__ void hyp_proj8(float* x) {
  float n = sqrtf(dot8(x, x));
  const float mx = 0.99999f;  // (1 - 1e-5)/sqrt(C)
  if (n > mx) {
    float s = mx / n;
#pragma unroll
    for (int j = 0; j < DD; ++j) x[j] *= s;
  }
}

__device__ __forceinline__ void hyp_exp8(const float* x, const float* v, float* o) {
  float nv = fmaxf(sqrtf(dot8(v, v)), 1e-10f);
  float nx2 = dot8(x, x);
  float lam = 2.f / (1.f - nx2);
  float coef = tanhf(lam * nv * 0.5f) / nv;
  float y[DD];
#pragma unroll
  for (int j = 0; j < DD; ++j) y[j] = coef * v[j];
  mobius_add8(x, y, o);
  hyp_proj8(o);
}

__device__ __forceinline__ void hyp_log8(const float* x, const float* y, float* o) {
  float nx_[DD], sub[DD];
#pragma unroll
  for (int j = 0; j < DD; ++j) nx_[j] = -x[j];
  mobius_add8(nx_, y, sub);
  float ns = fmaxf(sqrtf(dot8(sub, sub)), 1e-10f);
  float nx2 = dot8(x, x);
  float lam = 2.f / (1.f - nx2);
  float coef = atanhf(ns) / (ns * lam * 0.5f);
#pragma unroll
  for (int j = 0; j < DD; ++j) o[j] = coef * sub[j];
}

__device__ __forceinline__ void sph_proj8(float* x) {
  float s = 1.f / (sqrtf(dot8(x, x)) + 1e-10f);
#pragma unroll
  for (int j = 0; j < DD; ++j) x[j] *= s;
}

__device__ __forceinline__ void sph_log8(const float* x, const float* y, float* o) {
  float xy = dot8(x, y);
  xy = fminf(fmaxf(xy, -1.f + 1e-6f), 1.f - 1e-6f);
  float coef = acosf(xy) * rsqrtf(1.f - xy*xy + 1e-10f);
#pragma unroll
  for (int j = 0; j < DD; ++j) o[j] = coef * (y[j] - xy * x[j]);
}

__device__ __forceinline__ void sph_exp8(const float* x, const float* v, float* o) {
  float nv = fmaxf(sqrtf(dot8(v, v)), 1e-10f);
  float cn = cosf(nv), sn = sinf(nv) / nv;
#pragma unroll
  for (int j = 0; j < DD; ++j) o[j] = cn * x[j] + sn * v[j];
  sph_proj8(o);
}

// per-row "base" quantities shared by both passes
__device__ __forceinline__ void row_base(const float* mf,
    const float* W_hyp, const float* b_hyp,
    const float* W_spd, const float* b_spd,
    const float* W_sph, const float* b_sph,
    float* hyp_pt, float* spd_diag, float* sph_pt) {
  float zero8[DD] = {0,0,0,0,0,0,0,0};
  float ht[DD];
#pragma unroll
  for (int j = 0; j < DD; ++j) ht[j] = mf[0]*W_hyp[j] + mf[1]*W_hyp[DD + j] + b_hyp[j];
  hyp_exp8(zero8, ht, hyp_pt);
#pragma unroll
  for (int k = 0; k < 3; ++k)
    spd_diag[k] = expf(mf[0]*W_spd[k] + mf[1]*W_spd[3 + k] + mf[2]*W_spd[6 + k] + b_spd[k]) + 0.1f;
#pragma unroll
  for (int j = 0; j < DD; ++j) sph_pt[j] = mf[2]*W_sph[j] + mf[3]*W_sph[DD + j] + b_sph[j];
  sph_proj8(sph_pt);
}

// ---------------- kernel 0: zero the 19 accumulators ----------------
__global__ void rfe_zero_kernel(float* acc) {
  if (threadIdx.x < 19) acc[threadIdx.x] = 0.f;
}

// ---------------- kernel 1: global reductions ----------------
__global__ __launch_bounds__(256)
void rfe_stats_kernel(const float* __restrict__ feats,
                      const float* __restrict__ W_hyp, const float* __restrict__ b_hyp,
                      const float* __restrict__ W_spd, const float* __restrict__ b_spd,
                      const float* __restrict__ W_sph, const float* __restrict__ b_sph,
                      float* __restrict__ acc, long long N) {
  float s[19];
#pragma unroll
  for (int j = 0; j < 19; ++j) s[j] = 0.f;

  long long stride = (long long)gridDim.x * blockDim.x;
  for (long long r = (long long)blockIdx.x * blockDim.x + threadIdx.x; r < N; r += stride) {
    const float4* fp = (const float4*)(feats + r * 8);
    float4 f1 = fp[1];
    float mf[4] = {f1.x, f1.y, f1.z, f1.w};
    float hp[DD], sd[3], sp[DD];
    row_base(mf, W_hyp, b_hyp, W_spd, b_spd, W_sph, b_sph, hp, sd, sp);
#pragma unroll
    for (int j = 0; j < DD; ++j) s[j] += hp[j];
#pragma unroll
    for (int k = 0; k < 3; ++k) s[8 + k] += logf(sd[k] + 1e-8f);
#pragma unroll
    for (int j = 0; j < DD; ++j) s[11 + j] += sp[j];
  }

  // wave32 reduction
#pragma unroll
  for (int j = 0; j < 19; ++j) {
#pragma unroll
    for (int off = 16; off >= 1; off >>= 1) s[j] += __shfl_down(s[j], off, 32);
  }

  __shared__ float blk[19];
  if (threadIdx.x < 19) blk[threadIdx.x] = 0.f;
  __syncthreads();
  if ((threadIdx.x & 31) == 0) {
#pragma unroll
    for (int j = 0; j < 19; ++j) atomicAdd(&blk[j], s[j]);
  }
  __syncthreads();
  if (threadIdx.x < 19) atomicAdd(&acc[threadIdx.x], blk[threadIdx.x]);
}

// ---------------- kernel 2: finalize centers ----------------
__global__ void rfe_finalize_kernel(const float* __restrict__ acc, float* __restrict__ stats,
                                    long long N) {
  if (threadIdx.x != 0 || blockIdx.x != 0) return;
  float inv = 1.f / (float)N;
  float hc[DD];
#pragma unroll
  for (int j = 0; j < DD; ++j) hc[j] = acc[j] * inv;
  hyp_proj8(hc);
  float sc[DD];
#pragma unroll
  for (int j = 0; j < DD; ++j) sc[j] = acc[11 + j] * inv;
  sph_proj8(sc);
#pragma unroll
  for (int j = 0; j < DD; ++j) stats[j] = hc[j];
#pragma unroll
  for (int k = 0; k < 3; ++k) stats[8 + k] = expf(acc[8 + k] * inv);
#pragma unroll
  for (int j = 0; j < DD; ++j) stats[11 + j] = sc[j];
}

// ---------------- kernel 3: fused per-row + WMMA GEMM ----------------
#define RPB 128       // rows per block (= threads per block, 4 waves)
#define LSTR 36       // LDS row stride in floats:
                      //  - A-phase uses cols 0..23 (23 real + 1 zero pad)
                      //  - D-phase uses cols 0..31
                      //  - 36*4B = 144B row pitch: 16B aligned; 8*36 % 64 == 32
                      //    -> conflict-free D writes; 36 % 4 pattern -> conflict-free A reads

__global__ __launch_bounds__(RPB)
void rfe_fused_kernel(const float* __restrict__ feats,
                      const float* __restrict__ W_hyp, const float* __restrict__ b_hyp,
                      const float* __restrict__ W_spd, const float* __restrict__ b_spd,
                      const float* __restrict__ W_sph, const float* __restrict__ b_sph,
                      const float* __restrict__ W_int, const float* __restrict__ b_int,
                      const float* __restrict__ stats,
                      float* __restrict__ out, long long N) {
  __shared__ float tile[RPB * LSTR];

  const int tid = threadIdx.x;
  long long row = (long long)blockIdx.x * RPB + tid;
  bool valid = row < N;
  long long r = valid ? row : (N - 1);

  const float4* fp = (const float4*)(feats + r * 8);
  float4 f0 = fp[0];
  float4 f1 = fp[1];
  float mf[4] = {f1.x, f1.y, f1.z, f1.w};

  float hyp_pt[DD], spd_diag[3], sph_pt[DD];
  row_base(mf, W_hyp, b_hyp, W_spd, b_spd, W_sph, b_sph, hyp_pt, spd_diag, sph_pt);

  float hc[DD], gm[3], sc[DD];
#pragma unroll
  for (int j = 0; j < DD; ++j) hc[j] = stats[j];
#pragma unroll
  for (int k = 0; k < 3; ++k) gm[k] = stats[8 + k];
#pragma unroll
  for (int j = 0; j < DD; ++j) sc[j] = stats[11 + j];

  // hyp_ref = hyp_exp(hyp_pt, 0.1 * hyp_log(hyp_pt, hc))
  float hl[DD], hyp_ref[DD];
  hyp_log8(hyp_pt, hc, hl);
#pragma unroll
  for (int j = 0; j < DD; ++j) hl[j] *= 0.1f;
  hyp_exp8(hyp_pt, hl, hyp_ref);

  float spd_ref_diag[3];
#pragma unroll
  for (int k = 0; k < 3; ++k) spd_ref_diag[k] = fmaxf(0.9f * spd_diag[k] + 0.1f * gm[k], 1e-6f);

  float sl[DD], sph_ref[DD];
  sph_log8(sph_pt, sc, sl);
#pragma unroll
  for (int j = 0; j < DD; ++j) sl[j] *= 0.1f;
  sph_exp8(sph_pt, sl, sph_ref);

  float zero8[DD] = {0,0,0,0,0,0,0,0};
  float hyp_pull[DD];
  hyp_log8(zero8, hyp_ref, hyp_pull);

  float spd_pull[3];
#pragma unroll
  for (int k = 0; k < 3; ++k) spd_pull[k] = logf(spd_ref_diag[k] + 1e-8f);

  float comb[23];
  comb[0] = f0.x; comb[1] = f0.y; comb[2] = f0.z; comb[3] = f0.w;
#pragma unroll
  for (int j = 0; j < DD; ++j) comb[4 + j] = hyp_pull[j];
#pragma unroll
  for (int k = 0; k < 3; ++k) comb[12 + k] = spd_pull[k];
#pragma unroll
  for (int j = 0; j < DD; ++j) comb[15 + j] = sph_ref[j];

  // output regions (tuple flattened in return order)
  float* out_int = out;                         // (N,32)
  float* out_hyp = out + (size_t)N * 32;        // (N,8)
  float* out_spd = out + (size_t)N * 40;        // (N,3,3)
  float* out_sph = out + (size_t)N * 49;        // (N,8)
  float* out_cmb = out + (size_t)N * 57;        // (N,23)

  if (valid) {
    // hyp_ref, sph_ref: rows are 32B-aligned -> float4 stores
    float4* hv = (float4*)(out_hyp + row * 8);
    hv[0] = make_float4(hyp_ref[0], hyp_ref[1], hyp_ref[2], hyp_ref[3]);
    hv[1] = make_float4(hyp_ref[4], hyp_ref[5], hyp_ref[6], hyp_ref[7]);
    float4* sv = (float4*)(out_sph + row * 8);
    sv[0] = make_float4(sph_ref[0], sph_ref[1], sph_ref[2], sph_ref[3]);
    sv[1] = make_float4(sph_ref[4], sph_ref[5], sph_ref[6], sph_ref[7]);
    float* sd = out_spd + row * 9;
    sd[0] = spd_ref_diag[0]; sd[1] = 0.f; sd[2] = 0.f;
    sd[3] = 0.f; sd[4] = spd_ref_diag[1]; sd[5] = 0.f;
    sd[6] = 0.f; sd[7] = 0.f; sd[8] = spd_ref_diag[2];
    float* cm = out_cmb + row * 23;
#pragma unroll
    for (int k = 0; k < 23; ++k) cm[k] = comb[k];
  }

  // stage combined tile in LDS (K padded 23 -> 24 with zeros)
#pragma unroll
  for (int k = 0; k < 23; ++k) tile[tid * LSTR + k] = valid ? comb[k] : 0.f;
  tile[tid * LSTR + 23] = 0.f;
  __syncthreads();

  // ---- WMMA: integrated = relu(combined @ W_int + b_int), f32 16x16x4 ----
  const int wave = tid >> 5;
  const int lane = tid & 31;
  const int mrow = lane & 15;
  const int khalf = (lane >> 4) << 1;  // 0 for lanes 0-15, 2 for lanes 16-31
  const int moff = (lane >= 16) ? 8 : 0;

  // B fragments: chunk c covers K = 4c..4c+3; lane holds K = 4c+khalf, 4c+khalf+1
  v2f bfrag[2][6];
#pragma unroll
  for (int t = 0; t < 2; ++t) {
#pragma unroll
    for (int c = 0; c < 6; ++c) {
      int n = mrow + 16 * t;
      int k0 = 4 * c + khalf;
      bfrag[t][c].x = (k0     < 23) ? W_int[k0 * 32 + n]       : 0.f;
      bfrag[t][c].y = (k0 + 1 < 23) ? W_int[(k0 + 1) * 32 + n] : 0.f;
    }
  }

#pragma unroll
  for (int rt = 0; rt < 2; ++rt) {
    // each wave works entirely inside its own 32 rows: rows 32*wave .. 32*wave+31
    int tileRow = wave * 2 + rt;          // 16-row tile index
    v2f afrag[6];
#pragma unroll
    for (int c = 0; c < 6; ++c) {
      int k0 = 4 * c + khalf;
      const float* lp = &tile[(tileRow * 16 + mrow) * LSTR + k0];
      afrag[c].x = lp[0];
      afrag[c].y = lp[1];
    }
#pragma unroll
    for (int t = 0; t < 2; ++t) {
      v8f acc = {};
#pragma unroll
      for (int c = 0; c < 6; ++c) {
        acc = __builtin_amdgcn_wmma_f32_16x16x4_f32(
            false, afrag[c], false, bfrag[t][c], (short)0, acc, false, false);
      }
      int n = mrow + 16 * t;
      float bias = b_int[n];
      // write D tile (bias+relu applied) back into LDS, cols 0..31
#pragma unroll
      for (int rr = 0; rr < 8; ++rr) {
        tile[(tileRow * 16 + rr + moff) * LSTR + n] = fmaxf(acc[rr] + bias, 0.f);
      }
    }
  }
  __syncthreads();

  // each thread stores its own contiguous 32-float integrated row (b128 stores)
  if (valid) {
    float4* dst = (float4*)(out_int + row * 32);
    const float4* src = (const float4*)(&tile[tid * LSTR]);  // 144B pitch, 16B aligned
#pragma unroll
    for (int q = 0; q < 8; ++q) dst[q] = src[q];
  }
}

extern "C" void kernel_launch(void* const* d_in, const int* in_sizes, int n_in,
                              void* d_out, int out_size, void* d_ws, size_t ws_size,
                              hipStream_t stream) {
  const float* feats = (const float*)d_in[0];
  const float* W_hyp = (const float*)d_in[1];
  const float* b_hyp = (const float*)d_in[2];
  const float* W_spd = (const float*)d_in[3];
  const float* b_spd = (const float*)d_in[4];
  const float* W_sph = (const float*)d_in[5];
  const float* b_sph = (const float*)d_in[6];
  const float* W_int = (const float*)d_in[7];
  const float* b_int = (const float*)d_in[8];

  long long N = (long long)in_sizes[0] / 8;

  float* acc   = (float*)d_ws;   // 19 reduction accumulators
  float* stats = acc + 32;       // 19 finalized center values

  rfe_zero_kernel<<<1, 32, 0, stream>>>(acc);
  rfe_stats_kernel<<<256, 256, 0, stream>>>(feats, W_hyp, b_hyp, W_spd, b_spd,
                                            W_sph, b_sph, acc, N);
  rfe_finalize_kernel<<<1, 1, 0, stream>>>(acc, stats, N);

  long long nblk = (N + RPB - 1) / RPB;
  rfe_fused_kernel<<<dim3((unsigned)nblk), RPB, 0, stream>>>(
      feats, W_hyp, b_hyp, W_spd, b_spd, W_sph, b_sph, W_int, b_int,
      stats, (float*)d_out, N);
}